// MetaLSTM_10995116278126
// MI455X (gfx1250) — compile-verified
//
#include <hip/hip_runtime.h>

typedef float v2f __attribute__((ext_vector_type(2)));
typedef float v8f __attribute__((ext_vector_type(8)));

#define N_IN 20
#define KCHUNKS 5   // N_IN / 4 (K=4 per V_WMMA_F32_16X16X4_F32)

// sigmoid via raw v_rcp_f32: denominator is in [1, inf) so rcp is safe & ~1ulp
__device__ __forceinline__ float fast_sigmoid(float x) {
    return __builtin_amdgcn_rcpf(1.0f + __expf(-x));
}

__global__ __launch_bounds__(256) void metalstm_wmma_kernel(
    const float* __restrict__ x_all,  // [T, P, N_IN]
    const float* __restrict__ grad,   // [T]
    const float* __restrict__ WF,     // [N_IN+2]
    const float* __restrict__ WI,     // [N_IN+2]
    const float* __restrict__ cI,     // [P]
    const float* __restrict__ bF,     // [1]
    const float* __restrict__ bI,     // [1]
    float* __restrict__ out,          // [3*P] : c_T | f_T | i_T
    int P, int T)
{
    const int lane        = threadIdx.x & 31;          // wave32
    const int waveInBlock = threadIdx.x >> 5;
    const int wave        = blockIdx.x * (blockDim.x >> 5) + waveInBlock;
    const int p_base      = wave * 16;                 // 16 params per wave
    if (p_base >= P) return;                           // whole wave uniform exit

    const int m  = lane & 15;        // A-row index / B,D column (param) index
    const int kp = (lane >> 4) * 2;  // K-pair base within a 4-wide chunk

    // --- Build time-invariant A fragments: A[m][k] = (m==0)?WF[k] : (m==1)?WI[k] : 0
    v2f afrag[KCHUNKS];
    #pragma unroll
    for (int kc = 0; kc < KCHUNKS; ++kc) {
        float ax = 0.0f, ay = 0.0f;
        const int k0 = 4 * kc + kp;
        if (m == 0)      { ax = WF[k0]; ay = WF[k0 + 1]; }
        else if (m == 1) { ax = WI[k0]; ay = WI[k0 + 1]; }
        afrag[kc][0] = ax;
        afrag[kc][1] = ay;
    }

    // Recurrent-term weights and biases (uniform scalars)
    const float wf_c = WF[N_IN], wf_f = WF[N_IN + 1];
    const float wi_c = WI[N_IN], wi_i = WI[N_IN + 1];
    const float bfv  = bF[0],    biv  = bI[0];

    // Per-lane recurrence state for param p (lanes 0..15 own real params)
    const int p  = p_base + m;
    const int pc = (p < P) ? p : (P - 1);              // clamp for safe loads
    float c = cI[pc];
    float f = 0.0f;
    float i = 0.0f;

    const size_t stepStride = (size_t)P * N_IN;
    // Per-lane B-fragment base pointer: x[t][pc][kp]  (8-byte aligned: pc*20+kp is even)
    const float* xrow = x_all + (size_t)pc * N_IN + (size_t)kp;

    for (int t = 0; t < T; ++t) {
        // Prefetch next step's 1280B block (lanes collectively touch all 10 lines)
        if (t + 1 < T) __builtin_prefetch(xrow + stepStride, 0, 0);

        v8f acc = {};   // zero accumulator
        #pragma unroll
        for (int kc = 0; kc < KCHUNKS; ++kc) {
            // B (4x16): lane holds x[pc][4kc+kp], x[pc][4kc+kp+1]
            v2f b = *(const v2f*)(xrow + 4 * kc);
            // D = A x B + C   (exact f32 math)
            acc = __builtin_amdgcn_wmma_f32_16x16x4_f32(
                /*neg_a=*/false, afrag[kc],
                /*neg_b=*/false, b,
                /*c_mod=*/(short)0, acc,
                /*reuse_a=*/false, /*reuse_b=*/false);
        }

        // Lane l<16: acc[0] = x_t . WF[0:20]  for param p ;  acc[1] = x_t . WI[0:20]
        const float xf = acc[0];
        const float xi = acc[1];
        const float g  = grad[t];

        const float fn  = fmaf(c, wf_c, fmaf(f, wf_f, xf + bfv));
        const float in_ = fmaf(c, wi_c, fmaf(i, wi_i, xi + biv));
        const float sf  = fast_sigmoid(fn);
        const float si  = fast_sigmoid(in_);
        c = fmaf(sf, c, -si * g);
        f = fn;
        i = in_;

        xrow += stepStride;
    }

    // Only lanes 0..15 hold live params; outputs: c_T | f_T | i_T concatenated
    if (lane < 16 && p < P) {
        out[p]          = c;
        out[P + p]      = f;
        out[2 * P + p]  = i;
    }
}

extern "C" void kernel_launch(void* const* d_in, const int* in_sizes, int n_in,
                              void* d_out, int out_size, void* d_ws, size_t ws_size,
                              hipStream_t stream) {
    const float* x_all = (const float*)d_in[0];
    const float* grad  = (const float*)d_in[1];
    const float* WF    = (const float*)d_in[2];
    const float* WI    = (const float*)d_in[3];
    const float* cI    = (const float*)d_in[4];
    const float* bF    = (const float*)d_in[5];
    const float* bI    = (const float*)d_in[6];
    float* out = (float*)d_out;

    const int T = in_sizes[1];   // grad has one entry per step
    const int P = in_sizes[4];   // cI has one entry per param

    const int waves          = (P + 15) / 16;
    const int wavesPerBlock  = 8;                   // 256 threads = 8 wave32
    const int blocks         = (waves + wavesPerBlock - 1) / wavesPerBlock;

    metalstm_wmma_kernel<<<blocks, wavesPerBlock * 32, 0, stream>>>(
        x_all, grad, WF, WI, cI, bF, bI, out, P, T);
}